// Generator_63909113365210
// MI455X (gfx1250) — compile-verified
//
#include <hip/hip_runtime.h>

#define N_NODES 50000
#define N_EDGES 800000
#define BN_EPS 1e-5f

typedef __attribute__((ext_vector_type(2))) float v2f;
typedef __attribute__((ext_vector_type(8))) float v8f;

// ---------------------------------------------------------------- zero
__global__ void zero_kernel(float* __restrict__ p, int n) {
  int i = blockIdx.x * blockDim.x + threadIdx.x;
  if (i < n) p[i] = 0.f;
}

// ---------------------------------------------------------------- scatter-add aggregation
// agg[dst[e], :] += h[src[e], :]   ; one thread = one edge x 4 channels
template <int FI>
__global__ __launch_bounds__(256) void scatter_add_kernel(
    const float* __restrict__ h, const int* __restrict__ src,
    const int* __restrict__ dst, float* __restrict__ agg) {
  int e = blockIdx.x * blockDim.x + threadIdx.x;
  if (e >= N_EDGES) return;
  int c = blockIdx.y << 2;
  int s = src[e], d = dst[e];
  const float4 v = *(const float4*)(h + (size_t)s * FI + c);
  float* o = agg + (size_t)d * FI + c;
  atomicAdd(o + 0, v.x);
  atomicAdd(o + 1, v.y);
  atomicAdd(o + 2, v.z);
  atomicAdd(o + 3, v.w);
}

// ---------------------------------------------------------------- fused dual GEMM via WMMA f32
// out[m,n] = sum_k agg[m,k]*Wrel[k,n] + sum_k h[m,k]*Wroot[k,n] + bias[n]
// One wave32 per 16x16 output tile; K advanced 4 at a time with
// V_WMMA_F32_16X16X4_F32, fully unrolled (FI, FO compile-time).
// A layout (16x4 f32): lane m = lane&15; lanes<16 hold K=kh..kh+1 with kh=0,
// lanes>=16 kh=2. B mirrored. C/D: VGPR r <-> rows r / r+8 per lane half.
template <int FI, int FO>
__global__ __launch_bounds__(256) void gemm_dual_wmma(
    const float* __restrict__ agg, const float* __restrict__ h,
    const float* __restrict__ Wrel, const float* __restrict__ Wroot,
    const float* __restrict__ bias, float* __restrict__ out) {
  constexpr int NTN = (FO + 15) / 16;
  constexpr int TILES = (N_NODES / 16) * NTN;  // 50000 % 16 == 0
  constexpr bool GUARD = (FO % 16) != 0;

  int wave = (int)((blockIdx.x * blockDim.x + threadIdx.x) >> 5);
  int lane = threadIdx.x & 31;
  if (wave >= TILES) return;  // wave-uniform (EXEC stays full for WMMA)

  int mt = wave / NTN, nt = wave % NTN;
  int mrow = mt * 16 + (lane & 15);
  int col = nt * 16 + (lane & 15);
  bool colv = !GUARD || (col < FO);
  int colc = GUARD ? (colv ? col : 0) : col;  // clamped: loads always in-bounds
  int kh = (lane >> 4) << 1;                  // 0 or 2

  float bld = bias[colc];
  float bv = colv ? bld : 0.f;
  v8f acc = {bv, bv, bv, bv, bv, bv, bv, bv};

  const float* arow = agg + (size_t)mrow * FI;
  const float* hrow = h + (size_t)mrow * FI;
  const float* wr = Wrel + colc;
  const float* wo = Wroot + colc;

#pragma unroll
  for (int k = 0; k < FI; k += 4) {
    v2f a = *(const v2f*)(arow + k + kh);  // 8B-aligned (k+kh even)
    float w0 = wr[(k + kh) * FO];
    float w1 = wr[(k + kh + 1) * FO];
    v2f b;
    b.x = colv ? w0 : 0.f;  // cndmask, no EXEC branch (elided when !GUARD)
    b.y = colv ? w1 : 0.f;
    acc = __builtin_amdgcn_wmma_f32_16x16x4_f32(false, a, false, b, (short)0, acc,
                                                false, false);
  }
#pragma unroll
  for (int k = 0; k < FI; k += 4) {
    v2f a = *(const v2f*)(hrow + k + kh);
    float w0 = wo[(k + kh) * FO];
    float w1 = wo[(k + kh + 1) * FO];
    v2f b;
    b.x = colv ? w0 : 0.f;
    b.y = colv ? w1 : 0.f;
    acc = __builtin_amdgcn_wmma_f32_16x16x4_f32(false, a, false, b, (short)0, acc,
                                                false, false);
  }

  if (colv) {
    int rbase = mt * 16 + ((lane >> 4) << 3);  // rows 0..7 or 8..15 of tile
#pragma unroll
    for (int r = 0; r < 8; ++r)
      out[(size_t)(rbase + r) * FO + col] = acc[r];
  }
}

// ---------------------------------------------------------------- batchnorm statistics
// One block per channel: mean + rsqrt(biased_var + eps)
__global__ void bn_stats_kernel(const float* __restrict__ pre, int fo,
                                float* __restrict__ mean, float* __restrict__ rinv) {
  __shared__ float ssum[256];
  __shared__ float ssq[256];
  int ch = blockIdx.x;
  float s = 0.f, s2 = 0.f;
  for (int i = threadIdx.x; i < N_NODES; i += blockDim.x) {
    float v = pre[(size_t)i * fo + ch];
    s += v;
    s2 += v * v;
  }
  ssum[threadIdx.x] = s;
  ssq[threadIdx.x] = s2;
  __syncthreads();
  for (int off = 128; off > 0; off >>= 1) {
    if ((int)threadIdx.x < off) {
      ssum[threadIdx.x] += ssum[threadIdx.x + off];
      ssq[threadIdx.x] += ssq[threadIdx.x + off];
    }
    __syncthreads();
  }
  if (threadIdx.x == 0) {
    float mu = ssum[0] / (float)N_NODES;
    float var = ssq[0] / (float)N_NODES - mu * mu;  // biased var (training BN)
    mean[ch] = mu;
    rinv[ch] = rsqrtf(var + BN_EPS);
  }
}

// ---------------------------------------------------------------- batchnorm apply (+ReLU)
template <int FO, bool RELU>
__global__ __launch_bounds__(256) void bn_apply_kernel(
    const float* __restrict__ pre, const float* __restrict__ mean,
    const float* __restrict__ rinv, const float* __restrict__ gamma,
    const float* __restrict__ beta, float* __restrict__ out) {
  int idx = blockIdx.x * blockDim.x + threadIdx.x;
  constexpr int N = N_NODES * FO;
  if (idx >= N) return;
  int c = idx % FO;  // compile-time modulus
  float v = gamma[c] * (pre[idx] - mean[c]) * rinv[c] + beta[c];
  if (RELU) v = fmaxf(v, 0.f);
  out[idx] = v;
}

// ---------------------------------------------------------------- per-layer driver
template <int FI, int FO, bool RELU>
static void run_layer(const float* h, const int* src, const int* dst,
                      const float* Wrel, const float* Wroot, const float* bias,
                      const float* gamma, const float* beta, float* aggbuf,
                      float* pre, float* outp, float* meanb, float* rinvb,
                      hipStream_t stream) {
  // 1) zero aggregation buffer
  constexpr int NZ = N_NODES * FI;
  zero_kernel<<<(NZ + 255) / 256, 256, 0, stream>>>(aggbuf, NZ);

  // 2) scatter-add over edges
  dim3 sg((N_EDGES + 255) / 256, FI / 4);
  scatter_add_kernel<FI><<<sg, 256, 0, stream>>>(h, src, dst, aggbuf);

  // 3) fused dual GEMM + bias (WMMA f32), 8 waves / 256-thread block
  constexpr int TILES = (N_NODES / 16) * ((FO + 15) / 16);
  gemm_dual_wmma<FI, FO><<<(TILES + 7) / 8, 256, 0, stream>>>(aggbuf, h, Wrel,
                                                              Wroot, bias, pre);

  // 4) batch-norm stats
  bn_stats_kernel<<<FO, 256, 0, stream>>>(pre, FO, meanb, rinvb);

  // 5) normalize (+ReLU)
  constexpr int N = N_NODES * FO;
  bn_apply_kernel<FO, RELU><<<(N + 255) / 256, 256, 0, stream>>>(
      pre, meanb, rinvb, gamma, beta, outp);
}

// ---------------------------------------------------------------- orchestration
extern "C" void kernel_launch(void* const* d_in, const int* in_sizes, int n_in,
                              void* d_out, int out_size, void* d_ws, size_t ws_size,
                              hipStream_t stream) {
  (void)in_sizes; (void)n_in; (void)out_size; (void)ws_size;

  const float* x = (const float*)d_in[0];
  const int* ei = (const int*)d_in[1];
  const int* src = ei;            // edge_index[0]
  const int* dst = ei + N_EDGES;  // edge_index[1]

  const float *Wrel[5], *Wroot[5], *bvec[5], *gam[5], *bet[5];
  for (int i = 0; i < 5; ++i) {
    Wrel[i] = (const float*)d_in[2 + i];
    Wroot[i] = (const float*)d_in[7 + i];
    bvec[i] = (const float*)d_in[12 + i];
    gam[i] = (const float*)d_in[17 + i];
    bet[i] = (const float*)d_in[22 + i];
  }

  float* ws = (float*)d_ws;
  float* aggbuf = ws;                              // N*128 floats
  float* preA = aggbuf + (size_t)N_NODES * 128;    // N*128 floats
  float* preB = preA + (size_t)N_NODES * 128;      // N*128 floats
  float* meanb = preB + (size_t)N_NODES * 128;     // 128 floats
  float* rinvb = meanb + 128;                      // 128 floats

  // DIMS = [64, 64, 128, 32, 16, 3]; bn_apply is in-place for hidden layers
  // (pure elementwise), output buffer becomes next layer's input.
  run_layer<64, 64, true>(x, src, dst, Wrel[0], Wroot[0], bvec[0], gam[0], bet[0],
                          aggbuf, preA, preA, meanb, rinvb, stream);
  run_layer<64, 128, true>(preA, src, dst, Wrel[1], Wroot[1], bvec[1], gam[1], bet[1],
                           aggbuf, preB, preB, meanb, rinvb, stream);
  run_layer<128, 32, true>(preB, src, dst, Wrel[2], Wroot[2], bvec[2], gam[2], bet[2],
                           aggbuf, preA, preA, meanb, rinvb, stream);
  run_layer<32, 16, true>(preA, src, dst, Wrel[3], Wroot[3], bvec[3], gam[3], bet[3],
                          aggbuf, preB, preB, meanb, rinvb, stream);
  run_layer<16, 3, false>(preB, src, dst, Wrel[4], Wroot[4], bvec[4], gam[4], bet[4],
                          aggbuf, preA, (float*)d_out, meanb, rinvb, stream);
}